// LSTM_DIST_23502061044278
// MI455X (gfx1250) — compile-verified
//
#include <hip/hip_runtime.h>
#include <cstdint>
#include <cstddef>

// h(i,j) = [ fwd[j]-fwd[i-1] ; bwd[i]-bwd[j+1] ; fwd[i-1]-bwd[j+1] ]
// fwd[-1]=0, bwd[T]=0, rows with (i==0 && j==0) forced to all-zero.
// Shapes (reference): h (B=64, T=1024, H2=512) f32, spans (B,128,2) i32,
// out (B, 128, 768) f32.  Memory-bound gather: ~59MB of traffic.

typedef float float4v __attribute__((ext_vector_type(4)));
typedef int   v4i     __attribute__((ext_vector_type(4)));
typedef __attribute__((address_space(1))) v4i* g_v4i_ptr;   // global
typedef __attribute__((address_space(3))) v4i* l_v4i_ptr;   // LDS

#ifndef __has_builtin
#define __has_builtin(x) 0
#endif

constexpr int T_LEN = 1024;
constexpr int H2 = 512;
constexpr int H = 256;               // H2 / 2
constexpr int SPANS_PER_BLOCK = 4;   // 64 threads (2 waves) per span
constexpr int THREADS = 256;         // 8 wave32 per block

#if defined(__AMDGCN__) && __has_builtin(__builtin_amdgcn_global_load_async_to_lds_b128)
#define USE_ASYNC_LDS 1
#else
#define USE_ASYNC_LDS 0
#endif

__global__ __launch_bounds__(THREADS)
void lstm_minus_span_kernel(const float* __restrict__ h,
                            const int* __restrict__ spans,
                            float* __restrict__ out,
                            int n_spans, int batch_shift)
{
#if USE_ASYNC_LDS
    __shared__ float lds_fp[SPANS_PER_BLOCK][H];   // staged fwd[i-1] slices
    __shared__ float lds_bp[SPANS_PER_BLOCK][H];   // staged bwd[j+1] slices
#endif
    const int t  = threadIdx.x;
    const int sl = t >> 6;                 // span slot within block: 0..3
    const int p  = (t & 63) << 2;          // float offset in H: 0..252 step 4
    const int sg = blockIdx.x * SPANS_PER_BLOCK + sl;   // global span id
    if (sg >= n_spans) return;

    const int b = sg >> batch_shift;       // spans-per-batch is a power of two

    const int i = spans[2 * sg + 0];
    const int j = spans[2 * sg + 1];

    const bool pad        = (i == 0) && (j == 0);
    const bool pre_valid  = (i > 0);
    const bool post_valid = (j + 1 < T_LEN);
    const int  ipre  = pre_valid  ? (i - 1) : 0;           // clamped
    const int  jpost = post_valid ? (j + 1) : (T_LEN - 1); // clamped

    const float* rowbase = h + (size_t)b * T_LEN * H2;
    const float* fp_src = rowbase + (size_t)ipre  * H2 + p;       // fwd[i-1]
    const float* bp_src = rowbase + (size_t)jpost * H2 + H + p;   // bwd[j+1]

#if USE_ASYNC_LDS
    // CDNA5 async DMA global->LDS (ASYNCcnt-tracked), overlapped with the
    // two direct b128 loads below. Each thread later reads back exactly the
    // LDS bytes its own async op wrote, so a wave-local s_wait_asynccnt is
    // sufficient — no workgroup barrier required.
    {
        g_v4i_ptr gfp = (g_v4i_ptr)const_cast<float*>(fp_src);
        g_v4i_ptr gbp = (g_v4i_ptr)const_cast<float*>(bp_src);
        l_v4i_ptr lfp = (l_v4i_ptr)&lds_fp[sl][p];
        l_v4i_ptr lbp = (l_v4i_ptr)&lds_bp[sl][p];
        __builtin_amdgcn_global_load_async_to_lds_b128(gfp, lfp, 0, 0);
        __builtin_amdgcn_global_load_async_to_lds_b128(gbp, lbp, 0, 0);
    }
#endif

    // Direct b128 loads for the unconditionally-used rows (L2-resident h).
    const float4v fe = *(const float4v*)(rowbase + (size_t)j * H2 + p);      // fwd[j]
    const float4v bs = *(const float4v*)(rowbase + (size_t)i * H2 + H + p);  // bwd[i]

    const float4v vzero = {0.0f, 0.0f, 0.0f, 0.0f};
    float4v fp, bp;

#if USE_ASYNC_LDS
#if __has_builtin(__builtin_amdgcn_s_wait_asynccnt)
    __builtin_amdgcn_s_wait_asynccnt(0);
#else
    asm volatile("s_wait_asynccnt 0" ::: "memory");
#endif
    fp = *(const float4v*)&lds_fp[sl][p];
    bp = *(const float4v*)&lds_bp[sl][p];
#else
    fp = *(const float4v*)fp_src;
    bp = *(const float4v*)bp_src;
#endif

    if (!pre_valid)  fp = vzero;   // fwd[-1] = 0
    if (!post_valid) bp = vzero;   // bwd[T]  = 0

    float4v o0 = fe - fp;
    float4v o1 = bs - bp;
    float4v o2 = fp - bp;
    if (pad) { o0 = vzero; o1 = vzero; o2 = vzero; }

    // Output is streamed exactly once -> non-temporal stores keep h in L2.
    float* obase = out + (size_t)sg * (3 * H) + p;
    __builtin_nontemporal_store(o0, (float4v*)(obase));
    __builtin_nontemporal_store(o1, (float4v*)(obase + H));
    __builtin_nontemporal_store(o2, (float4v*)(obase + 2 * H));
}

extern "C" void kernel_launch(void* const* d_in, const int* in_sizes, int n_in,
                              void* d_out, int out_size, void* d_ws, size_t ws_size,
                              hipStream_t stream) {
    const float* h     = (const float*)d_in[0];  // (B, T, H2) f32
    const int*   spans = (const int*)d_in[1];    // (B, S, 2)  i32
    float*       out   = (float*)d_out;          // (B, S, 3H) f32

    const int n_spans         = in_sizes[1] / 2;            // B*S
    const int B               = in_sizes[0] / (T_LEN * H2); // 64
    const int spans_per_batch = (B > 0) ? (n_spans / B) : 1; // S = 128 (pow2)

    int batch_shift = 0;
    while ((1 << batch_shift) < spans_per_batch) ++batch_shift;  // log2(S) = 7

    const int blocks = (n_spans + SPANS_PER_BLOCK - 1) / SPANS_PER_BLOCK;
    lstm_minus_span_kernel<<<blocks, THREADS, 0, stream>>>(
        h, spans, out, n_spans, batch_shift);
}